// AFM_75273596829798
// MI455X (gfx1250) — compile-verified
//
#include <hip/hip_runtime.h>
#include <math.h>

// ---------------------------------------------------------------------------
// AFM (attention factorization machine) fused kernel for MI455X / gfx1250.
// One workgroup (128 threads = 4 wave32) per batch sample.
// inner[276,16] @ attn_W[16,32] via V_WMMA_F32_16X16X4_F32:
//   - each wave owns M-tiles (16 pair-rows), strided by 4 waves
//   - per M-tile: both N-tiles computed (2 accumulators, 8 chained WMMAs)
//   - B fragments (2 N-tiles x 4 K-steps) hoisted into registers per wave
//   - A fragments loaded unconditionally from LDS (ds_load_2addr_b64),
//     padding rows masked arithmetically (no exec-mask branches)
//   - scoring: per-lane column partials stored straight to LDS (no
//     bpermute butterflies, no waits in the hot loop); 16-wide column
//     sum folded into the softmax pass as ds_load_b128 + adds.
// ---------------------------------------------------------------------------

typedef __attribute__((ext_vector_type(2))) float v2f;
typedef __attribute__((ext_vector_type(8))) float v8f;

#define NF        24          // fields
#define FIELD_DIM 50000
#define E         16          // embed dim
#define ATT       32          // attn size
#define NP        276         // NF*(NF-1)/2 pairs
#define NPP       288         // padded to 18 tiles of 16
#define MT        18          // M tiles
#define BLK       128         // 4 wave32

// compile-time pair (row,col) tables, padded to 288
struct PairTab { unsigned char r[NPP]; unsigned char c[NPP]; };
constexpr PairTab make_pairs() {
    PairTab t{};
    int p = 0;
    for (int i = 0; i < NF; ++i)
        for (int j = i + 1; j < NF; ++j) { t.r[p] = (unsigned char)i; t.c[p] = (unsigned char)j; ++p; }
    for (; p < NPP; ++p) { t.r[p] = 0; t.c[p] = 0; }
    return t;
}
__constant__ PairTab PAIRS = make_pairs();

__global__ __launch_bounds__(BLK) void afm_fused_kernel(
    const int*   __restrict__ x,            // [B, NF] int32
    const float* __restrict__ embed_table,  // [VOCAB, E]
    const float* __restrict__ linear_table, // [VOCAB]
    const float* __restrict__ linear_bias,  // [1]
    const float* __restrict__ attn_W,       // [E, ATT]
    const float* __restrict__ attn_b,       // [ATT]
    const float* __restrict__ proj_w,       // [ATT]
    const float* __restrict__ proj_b,       // [1]
    const float* __restrict__ fc_w,         // [E]
    const float* __restrict__ fc_b,         // [1]
    float*       __restrict__ out)          // [B]
{
    __shared__ float s_emb[NF][E];          // gathered embeddings
    __shared__ float s_W[E][ATT];           // attn_W
    __shared__ float s_ab[ATT];             // attn_b
    __shared__ float s_pw[ATT];             // proj_w
    __shared__ float s_part[NPP][16];       // per-(row, col-lane) score partials
    __shared__ float s_scores[NPP];         // scores -> exp(scores)
    __shared__ float s_lin[NF];             // per-field linear terms
    __shared__ float s_red[BLK];            // reduction scratch
    __shared__ float s_pool[8][E];          // pooled partials
    __shared__ unsigned char s_rf[NPP];     // pair row field
    __shared__ unsigned char s_cf[NPP];     // pair col field

    const int b = blockIdx.x;
    const int t = threadIdx.x;

    // ---------------- Phase 1: gathers + staging into LDS ----------------
    if (t < NF * 4) {                       // 96 threads, one float4 each
        const int f = t >> 2, q = t & 3;
        const size_t row = (size_t)x[b * NF + f] + (size_t)f * FIELD_DIM;
        const float4* src = (const float4*)(embed_table + row * E);
        ((float4*)&s_emb[f][0])[q] = src[q];
    }
    if (t < NF) {
        const size_t row = (size_t)x[b * NF + t] + (size_t)t * FIELD_DIM;
        s_lin[t] = linear_table[row];
    }
    for (int i = t; i < E * ATT; i += BLK) ((float*)s_W)[i] = attn_W[i];
    if (t < ATT) { s_ab[t] = attn_b[t]; s_pw[t] = proj_w[t]; }
    for (int i = t; i < NPP; i += BLK) { s_rf[i] = PAIRS.r[i]; s_cf[i] = PAIRS.c[i]; }
    const float pb = proj_b[0];             // uniform scalar load
    __syncthreads();

    // ---------------- Phase 2: WMMA MLP + partial scoring ----------------
    const int wave = t >> 5;                // wave32
    const int lane = t & 31;
    const int n    = lane & 15;             // A row / B,C col within tile
    const int hi   = lane >> 4;             // K-half select (ISA A/B 16x4 layout)

    // Hoist B fragments: depend only on (nt, kk, lane). 8 x v2f in registers.
    v2f bf[2][4];
    #pragma unroll
    for (int nt = 0; nt < 2; ++nt) {
        const int col = (nt << 4) + n;
        #pragma unroll
        for (int kk = 0; kk < 4; ++kk) {
            const int k = (kk << 2) + (hi << 1);
            bf[nt][kk].x = s_W[k][col];
            bf[nt][kk].y = s_W[k + 1][col];
        }
    }
    const float ab0 = s_ab[n],      pw0 = s_pw[n];
    const float ab1 = s_ab[16 + n], pw1 = s_pw[16 + n];

    for (int mtile = wave; mtile < MT; mtile += 4) {
        const int p = (mtile << 4) + n;     // this lane's A-matrix row (pair id)
        const float msk = (p < NP) ? 1.f : 0.f;
        const float* ra = &s_emb[s_rf[p]][0];
        const float* rc = &s_emb[s_cf[p]][0];

        // A fragments: unconditional LDS loads, masked arithmetically.
        v2f a[4];
        #pragma unroll
        for (int kk = 0; kk < 4; ++kk) {
            const int k = (kk << 2) + (hi << 1);
            a[kk].x = ra[k]     * rc[k]     * msk;
            a[kk].y = ra[k + 1] * rc[k + 1] * msk;
        }

        v8f c0 = {0.f,0.f,0.f,0.f,0.f,0.f,0.f,0.f};
        v8f c1 = {0.f,0.f,0.f,0.f,0.f,0.f,0.f,0.f};
        #pragma unroll
        for (int kk = 0; kk < 4; ++kk) {    // interleave the two accumulator chains
            c0 = __builtin_amdgcn_wmma_f32_16x16x4_f32(
                    false, a[kk], false, bf[0][kk], (short)0, c0, false, false);
            c1 = __builtin_amdgcn_wmma_f32_16x16x4_f32(
                    false, a[kk], false, bf[1][kk], (short)0, c1, false, false);
        }

        // bias + ReLU + proj weight for this lane's two attn columns; store
        // the per-column partial straight to LDS (padded rows included ->
        // no branches, no in-loop waits; column sum happens in phase 3).
        #pragma unroll
        for (int r = 0; r < 8; ++r) {
            const float h0 = fmaxf(c0[r] + ab0, 0.f);
            const float h1 = fmaxf(c1[r] + ab1, 0.f);
            const int prow = (mtile << 4) + r + (hi << 3);  // C/D layout row
            s_part[prow][n] = h0 * pw0 + h1 * pw1;
        }
    }
    __syncthreads();

    // ---------------- Phase 3: column-sum + softmax over 276 pairs ----------------
    float lmax = -3.4e38f;
    for (int p = t; p < NP; p += BLK) {
        const float4* rp = (const float4*)&s_part[p][0];
        const float4 q0 = rp[0], q1 = rp[1], q2 = rp[2], q3 = rp[3];
        const float s = pb
            + (((q0.x + q0.y) + (q0.z + q0.w)) + ((q1.x + q1.y) + (q1.z + q1.w)))
            + (((q2.x + q2.y) + (q2.z + q2.w)) + ((q3.x + q3.y) + (q3.z + q3.w)));
        s_scores[p] = s;
        lmax = fmaxf(lmax, s);
    }
    s_red[t] = lmax;
    __syncthreads();
    for (int s = BLK / 2; s > 0; s >>= 1) {
        if (t < s) s_red[t] = fmaxf(s_red[t], s_red[t + s]);
        __syncthreads();
    }
    const float mx = s_red[0];
    __syncthreads();

    float lsum = 0.f;
    for (int p = t; p < NP; p += BLK) {
        const float e0 = __expf(s_scores[p] - mx);
        s_scores[p] = e0;
        lsum += e0;
    }
    s_red[t] = lsum;
    __syncthreads();
    for (int s = BLK / 2; s > 0; s >>= 1) {
        if (t < s) s_red[t] += s_red[t + s];
        __syncthreads();
    }
    const float inv = 1.f / s_red[0];
    __syncthreads();

    // ---------------- Phase 4: attention pooling (recompute inner) ----------------
    const int e  = t & 15;                  // embed lane
    const int pg = t >> 4;                  // 0..7 pair groups
    float acc = 0.f;
    for (int p = pg; p < NP; p += 8) {
        const float in = s_emb[s_rf[p]][e] * s_emb[s_cf[p]][e];
        acc += s_scores[p] * in;            // s_scores holds exp()
    }
    s_pool[pg][e] = acc;
    __syncthreads();

    if (t < E) {
        float pooled = 0.f;
        #pragma unroll
        for (int g = 0; g < 8; ++g) pooled += s_pool[g][t];
        pooled *= inv;                      // softmax normalization folded here
        s_red[t] = pooled * fc_w[t];
    }
    __syncthreads();

    if (t == 0) {
        float cross = 0.f;
        #pragma unroll
        for (int i = 0; i < E; ++i) cross += s_red[i];
        float lin = linear_bias[0];
        #pragma unroll
        for (int f = 0; f < NF; ++f) lin += s_lin[f];
        out[b] = lin + cross + fc_b[0];
    }
}

extern "C" void kernel_launch(void* const* d_in, const int* in_sizes, int n_in,
                              void* d_out, int out_size, void* d_ws, size_t ws_size,
                              hipStream_t stream) {
    (void)n_in; (void)d_ws; (void)ws_size; (void)out_size;
    const int*   x            = (const int*)  d_in[0];
    const float* embed_table  = (const float*)d_in[1];
    const float* linear_table = (const float*)d_in[2];
    const float* linear_bias  = (const float*)d_in[3];
    const float* attn_W       = (const float*)d_in[4];
    const float* attn_b       = (const float*)d_in[5];
    const float* proj_w       = (const float*)d_in[6];
    const float* proj_b       = (const float*)d_in[7];
    const float* fc_w         = (const float*)d_in[8];
    const float* fc_b         = (const float*)d_in[9];
    float*       out          = (float*)d_out;

    const int B = in_sizes[0] / NF;         // 16384
    afm_fused_kernel<<<B, BLK, 0, stream>>>(
        x, embed_table, linear_table, linear_bias,
        attn_W, attn_b, proj_w, proj_b, fc_w, fc_b, out);
}